// InversePixelShuffle_35845797053126
// MI455X (gfx1250) — compile-verified
//
#include <hip/hip_runtime.h>

// Pixel-unshuffle k=2 on fp32 (8, 32, 512, 512) -> (8, 128, 256, 256).
// Pure bandwidth problem: 537 MB total traffic, ~23 us at 23.3 TB/s.
// Strategy: 8 input floats per thread (2x b128 NT loads), register shuffle,
// 2x b128 NT stores into the two target output channels. All accesses
// coalesced; indexing is shifts/masks only.

typedef __attribute__((ext_vector_type(4))) float float4v;

namespace {
constexpr unsigned kB  = 8;
constexpr unsigned kC  = 32;
constexpr unsigned kH  = 512;
constexpr unsigned kW  = 512;
constexpr unsigned kHo = kH / 2;          // 256
constexpr unsigned kWo = kW / 2;          // 256
constexpr unsigned kChunk        = 8;     // input floats per thread
constexpr unsigned kChunksPerRow = kW / kChunk;               // 64
constexpr unsigned kTotalChunks  = kB * kC * kH * kChunksPerRow; // 8,388,608
constexpr unsigned kBlock = 256;          // 8 wave32s per block
}

__global__ __launch_bounds__(kBlock)
void pixel_unshuffle2_f32_kernel(const float* __restrict__ in,
                                 float* __restrict__ out) {
    const unsigned t = blockIdx.x * kBlock + threadIdx.x;   // < 2^23, exact grid

    const unsigned j    = t & (kChunksPerRow - 1);  // chunk index within input row
    const unsigned row  = t >> 6;                   // flat input row = (b*C + c)*H + h_in
    const unsigned h_in = row & (kH - 1);
    const unsigned bc   = row >> 9;                 // b*C + c
    const unsigned h    = h_in >> 1;
    const unsigned y    = h_in & 1;

    // ---- load 8 contiguous input floats (streaming, read-once) ----
    const float* src = in + (size_t)row * kW + (size_t)j * kChunk;
    const float4v v0 = __builtin_nontemporal_load((const float4v*)src);
    const float4v v1 = __builtin_nontemporal_load((const float4v*)(src + 4));

    // even w -> out channel 4*bc + 2y ; odd w -> 4*bc + 2y + 1
    const float4v ev = __builtin_shufflevector(v0, v1, 0, 2, 4, 6);
    const float4v od = __builtin_shufflevector(v0, v1, 1, 3, 5, 7);

    // ---- store two contiguous float4s (streaming, write-once) ----
    // out index: ((4*bc + 2*y + parity) * Ho + h) * Wo + 4*j
    const size_t obase = ((size_t)(4u * bc + 2u * y) * kHo + h) * kWo + (size_t)j * 4u;
    __builtin_nontemporal_store(ev, (float4v*)(out + obase));
    __builtin_nontemporal_store(od, (float4v*)(out + obase + (size_t)kHo * kWo));
}

extern "C" void kernel_launch(void* const* d_in, const int* in_sizes, int n_in,
                              void* d_out, int out_size, void* d_ws, size_t ws_size,
                              hipStream_t stream) {
    (void)in_sizes; (void)n_in; (void)out_size; (void)d_ws; (void)ws_size;
    const float* in = (const float*)d_in[0];
    float* out = (float*)d_out;

    const unsigned grid = kTotalChunks / kBlock;  // 32768 blocks, exact cover
    pixel_unshuffle2_f32_kernel<<<grid, kBlock, 0, stream>>>(in, out);
}